// radarLoss_24343874634355
// MI455X (gfx1250) — compile-verified
//
#include <hip/hip_runtime.h>
#include <hip/hip_bf16.h>
#include <math.h>

// ---------------------------------------------------------------------------
// Radar loss, fused: per 16-batch tile compute R[b,d], I[b,d] (d = |a-m|)
// with V_WMMA_F32_16X16X4_F32, then stream y_true once and reduce RMSE.
// Memory-bound: ~233 MB traffic -> ~10us floor at 23.3 TB/s.
// ---------------------------------------------------------------------------

typedef float v2f __attribute__((ext_vector_type(2)));
typedef float v8f __attribute__((ext_vector_type(8)));

#define N_ANGLES   121
#define KPAD       128
#define NSTEPS     (KPAD / 4)   // 32 WMMA K-steps
#define TILE_B     16
#define WAVES_WG   4
#define BLOCK      (WAVES_WG * 32)
#define BATCH      65536
#define N_TILES    (BATCH / TILE_B)          // 4096
#define NWG        (N_TILES / WAVES_WG)      // 1024
#define ELEMS_PER_TILE (TILE_B * 3 * 16 * 16)   // 12288 floats
#define Q_PER_TILE (ELEMS_PER_TILE / 4)         // 3072 float4
#define PI_F 3.14159265358979323846f

// ---------------------------------------------------------------------------
// Init: build WMMA B-fragments for the cos/sin steering tables.
// Layout: btab[t*64 + l*2 + r]  (t = K-step, l = lane, r = B-frag VGPR index)
//   K = 4t + r + 2*(l>=16), N(=d) = l & 15
//   cos table at [0, 2048), sin table at [2048, 4096)
// ---------------------------------------------------------------------------
__global__ void radar_init_btab(float* __restrict__ btab) {
    int idx = blockIdx.x * blockDim.x + threadIdx.x;
    if (idx >= 2048) return;
    int t   = idx >> 6;
    int rem = idx & 63;
    int l   = rem >> 1;
    int r   = rem & 1;
    int K   = t * 4 + r + ((l >> 4) << 1);
    int N   = l & 15;
    float cv = 0.0f, sv = 0.0f;
    if (K < N_ANGLES) {
        float ang = (-60.0f + (float)K) * (PI_F / 180.0f);
        float ph  = PI_F * (float)N * __sinf(ang);
        cv = cosf(ph);
        sv = sinf(ph);
    }
    btab[idx]        = cv;
    btab[2048 + idx] = sv;
}

// ---------------------------------------------------------------------------
// Main fused kernel: one 16-batch tile per wave32.
// ---------------------------------------------------------------------------
__global__ __launch_bounds__(BLOCK)
void radar_loss_main(const float* __restrict__ y_pred,   // [B,121]
                     const float* __restrict__ y_true,   // [B,3,16,16]
                     const float* __restrict__ btab,     // [2][32][32][2]
                     float* __restrict__ partials) {     // [NWG]
    __shared__ float pLds[WAVES_WG][TILE_B * KPAD];   // 4 * 8KB
    __shared__ float rLds[WAVES_WG][TILE_B][16];      // 4 * 1KB
    __shared__ float iLds[WAVES_WG][TILE_B][16];      // 4 * 1KB
    __shared__ float red[BLOCK];

    const int tid  = threadIdx.x;
    const int wave = tid >> 5;
    const int lane = tid & 31;
    const int tileId = blockIdx.x * WAVES_WG + wave;
    const long b0 = (long)tileId * TILE_B;

    // ---- 1. stage p-tile into LDS, zero-padded K: 121 -> 128 --------------
    const float* prow = y_pred + b0 * N_ANGLES;
    for (int j = lane; j < TILE_B * KPAD; j += 32) {
        int bt = j >> 7;
        int n  = j & (KPAD - 1);
        pLds[wave][j] = (n < N_ANGLES) ? prow[bt * N_ANGLES + n] : 0.0f;
    }
    __syncthreads();

    // ---- 2. [16 x 128] x [128 x 16] GEMMs (cos & sin) via WMMA f32 --------
    // A frag (16x4 f32): VGPR r: lane<16 -> (M=lane, K=4t+r),
    //                            lane>=16 -> (M=lane-16, K=4t+2+r)
    v8f accR = {};
    v8f accI = {};
    const int   mrow  = lane & 15;
    const int   khalf = (lane >> 4) << 1;           // 0 or 2
    const float* cB = btab + lane * 2;              // + t*64 per step
    const float* sB = btab + 2048 + lane * 2;

    for (int t = 0; t < NSTEPS; ++t) {
        v2f afrag = *(const v2f*)&pLds[wave][mrow * KPAD + t * 4 + khalf];
        v2f bc    = *(const v2f*)&cB[t * 64];
        v2f bs    = *(const v2f*)&sB[t * 64];
        accR = __builtin_amdgcn_wmma_f32_16x16x4_f32(
                   false, afrag, false, bc, (short)0, accR, false, false);
        accI = __builtin_amdgcn_wmma_f32_16x16x4_f32(
                   false, afrag, false, bs, (short)0, accI, false, false);
    }

    // ---- 3. D-matrix (16x16 f32) -> LDS: VGPR r holds M=r (lanes 0-15),
    //         M=r+8 (lanes 16-31); N = lane & 15.  M = batch row, N = d.
    {
        const int mhi = (lane >> 4) << 3;           // 0 or 8
        const int nn  = lane & 15;
#pragma unroll
        for (int r = 0; r < 8; ++r) {
            rLds[wave][r + mhi][nn] = accR[r];
            iLds[wave][r + mhi][nn] = accI[r];
        }
    }
    __syncthreads();

    // ---- 4. stream y_true for this tile once; accumulate squared error ----
    float acc = 0.0f;
    const float4* yt = (const float4*)(y_true + b0 * 768);
    for (int q = lane; q < Q_PER_TILE; q += 32) {
        float4 v = yt[q];
        int b_local = q / 192;               // 192 float4 per batch
        int rem     = q - b_local * 192;
        int c       = rem / 64;              // channel: 0=re 1=im 2=ang
        int w       = rem - c * 64;          // float4 within 16x16
        int a       = w >> 2;
        int mb      = (w & 3) << 2;
        const float* Rr = rLds[wave][b_local];
        const float* Ii = iLds[wave][b_local];
        float vv[4] = { v.x, v.y, v.z, v.w };
#pragma unroll
        for (int e = 0; e < 4; ++e) {
            int mm = mb + e;
            int dd = a - mm;
            int ad = dd < 0 ? -dd : dd;
            float re = Rr[ad];
            float im = dd < 0 ? -Ii[ad] : Ii[ad];
            float pred = (c == 0) ? re : (c == 1) ? im : atan2f(im, re);
            float df = vv[e] - pred;
            acc = fmaf(df, df, acc);
        }
    }

    // ---- 5. deterministic block reduction -> one partial per WG -----------
    red[tid] = acc;
    __syncthreads();
    for (int s = BLOCK / 2; s > 0; s >>= 1) {
        if (tid < s) red[tid] += red[tid + s];
        __syncthreads();
    }
    if (tid == 0) partials[blockIdx.x] = red[0];
}

// ---------------------------------------------------------------------------
// Final reduction: sum partials in fixed order, sqrt(mean).
// ---------------------------------------------------------------------------
__global__ __launch_bounds__(256)
void radar_loss_final(const float* __restrict__ partials, int n,
                      float* __restrict__ out) {
    __shared__ float red[256];
    float a = 0.0f;
    for (int i = threadIdx.x; i < n; i += 256) a += partials[i];
    red[threadIdx.x] = a;
    __syncthreads();
    for (int s = 128; s > 0; s >>= 1) {
        if (threadIdx.x < s) red[threadIdx.x] += red[threadIdx.x + s];
        __syncthreads();
    }
    if (threadIdx.x == 0) {
        const float inv_count = 1.0f / (65536.0f * 3.0f * 16.0f * 16.0f);
        out[0] = sqrtf(red[0] * inv_count);
    }
}

// ---------------------------------------------------------------------------
extern "C" void kernel_launch(void* const* d_in, const int* in_sizes, int n_in,
                              void* d_out, int out_size, void* d_ws, size_t ws_size,
                              hipStream_t stream) {
    const float* y_pred = (const float*)d_in[0];   // [65536, 121] f32
    const float* y_true = (const float*)d_in[1];   // [65536, 3, 16, 16] f32
    float* out = (float*)d_out;                    // scalar f32

    float* btab     = (float*)d_ws;                // 4096 floats (16 KB)
    float* partials = btab + 4096;                 // NWG floats

    radar_init_btab<<<8, 256, 0, stream>>>(btab);
    radar_loss_main<<<NWG, BLOCK, 0, stream>>>(y_pred, y_true, btab, partials);
    radar_loss_final<<<1, 256, 0, stream>>>(partials, NWG, out);
}